// AttnDecoderRNN_7567732376035
// MI455X (gfx1250) — compile-verified
//
#include <hip/hip_runtime.h>
#include <hip/hip_bf16.h>

// Sizes fixed by the reference
#define V 32000
#define E 512
#define H 1024
#define NB 64
#define S 2048

typedef __attribute__((ext_vector_type(2))) float v2f;
typedef __attribute__((ext_vector_type(8))) float v8f;

// ---------------------------------------------------------------------------
// Zero-init a float buffer (concat scratch: atomics accumulate into it)
// ---------------------------------------------------------------------------
__global__ void k_zero(float* p, int n) {
    int i = blockIdx.x * blockDim.x + threadIdx.x;
    if (i < n) p[i] = 0.0f;
}

// ---------------------------------------------------------------------------
// gates[n][j] = sum_k emb[idx[n]][k]*w_ih[j][k] + sum_k h0[n][k]*w_hh[j][k]
//              + b_ih[j] + b_hh[j]
// One wave per 16-col tile jt (256 waves). Each wave holds 4 accumulators
// covering all M=64 rows, so w_ih/w_hh B-fragments are loaded ONCE and fed
// to 4 WMMAs (B reuse in registers instead of 4x global traffic).
// WMMA f32 16x16x4 layouts:
//   A: lanes 0-15 -> K={kb,kb+1}, lanes 16-31 -> K={kb+2,kb+3}, M = lane&15
//   B: lane holds N=lane&15, same K split
//   C/D: VGPR r: lanes 0-15 -> M=r, lanes 16-31 -> M=r+8; N = lane&15
// ---------------------------------------------------------------------------
__global__ void k_gates(const int* __restrict__ idx,
                        const float* __restrict__ h0,
                        const float* __restrict__ embW,
                        const float* __restrict__ w_ih,
                        const float* __restrict__ w_hh,
                        const float* __restrict__ b_ih,
                        const float* __restrict__ b_hh,
                        float* __restrict__ gates) {
    const int jt    = blockIdx.x * (blockDim.x >> 5) + (threadIdx.x >> 5); // 0..255
    const int lane  = threadIdx.x & 31;
    const int mrow  = lane & 15;
    const int khalf = (lane >> 4) << 1;   // 0 or 2

    const int j = jt * 16 + mrow;         // B-col this lane feeds
    const float* birow = w_ih + (size_t)j * E;
    const float* bhrow = w_hh + (size_t)j * H;

    // A row pointers for the 4 row tiles (mt = 0..3)
    const float* arow[4];
    const float* hrow[4];
    #pragma unroll
    for (int mt = 0; mt < 4; ++mt) {
        int n = mt * 16 + mrow;
        arow[mt] = embW + (size_t)idx[n] * E;
        hrow[mt] = h0   + (size_t)n * H;
    }

    v8f acc[4] = {v8f{}, v8f{}, v8f{}, v8f{}};

    for (int k = 0; k < E; k += 4) {
        v2f b;
        b.x = birow[k + khalf]; b.y = birow[k + khalf + 1];
        #pragma unroll
        for (int mt = 0; mt < 4; ++mt) {
            v2f a;
            a.x = arow[mt][k + khalf]; a.y = arow[mt][k + khalf + 1];
            acc[mt] = __builtin_amdgcn_wmma_f32_16x16x4_f32(
                false, a, false, b, (short)0, acc[mt], false, false);
        }
    }
    for (int k = 0; k < H; k += 4) {
        v2f b;
        b.x = bhrow[k + khalf]; b.y = bhrow[k + khalf + 1];
        #pragma unroll
        for (int mt = 0; mt < 4; ++mt) {
            v2f a;
            a.x = hrow[mt][k + khalf]; a.y = hrow[mt][k + khalf + 1];
            acc[mt] = __builtin_amdgcn_wmma_f32_16x16x4_f32(
                false, a, false, b, (short)0, acc[mt], false, false);
        }
    }

    const int   colN = jt * 16 + (lane & 15);
    const float bias = b_ih[colN] + b_hh[colN];
    const int   mofs = (lane >> 4) * 8;
    #pragma unroll
    for (int mt = 0; mt < 4; ++mt) {
        #pragma unroll
        for (int r = 0; r < 8; ++r) {
            int row = mt * 16 + r + mofs;
            gates[(size_t)row * (4 * H) + colN] = acc[mt][r] + bias;
        }
    }
}

// ---------------------------------------------------------------------------
// LSTM cell elementwise. Writes h,c to d_out regions and h into concat[:, :H].
// ---------------------------------------------------------------------------
__global__ void k_lstm(const float* __restrict__ gates,
                       const float* __restrict__ c0,
                       float* __restrict__ h_out,
                       float* __restrict__ c_out,
                       float* __restrict__ concat) {
    int t = blockIdx.x * blockDim.x + threadIdx.x;   // 0 .. 64*1024-1
    if (t >= NB * H) return;
    int n = t >> 10, hh = t & (H - 1);
    const float* g = gates + (size_t)n * (4 * H);
    float gi = g[hh], gf = g[H + hh], gg = g[2 * H + hh], go = g[3 * H + hh];
    float si = 1.0f / (1.0f + __expf(-gi));
    float sf = 1.0f / (1.0f + __expf(-gf));
    float so = 1.0f / (1.0f + __expf(-go));
    float c  = sf * c0[t] + si * tanhf(gg);
    float h  = so * tanhf(c);
    h_out[t] = h;
    c_out[t] = c;
    concat[(size_t)n * (2 * H) + hh] = h;
}

// ---------------------------------------------------------------------------
// Fused attention: single pass over enc.
//   score_s    = dot(enc[n,s,:], h[n])
//   context[n]+= score_s * enc[n,s,:]
// Block = (n, 64-row chunk of s). enc row cached in regs, h in LDS,
// partial context in regs, atomicAdd into concat[:, H:2H].
// ---------------------------------------------------------------------------
__global__ void k_attn(const float* __restrict__ enc,
                       float* __restrict__ concat) {
    const int n  = blockIdx.x;    // 0..63
    const int sc = blockIdx.y;    // 0..31 (chunks of 64 rows)
    const int t  = threadIdx.x;   // 0..255

    __shared__ float h_s[H];
    __shared__ float wsum[8];

    const float* hrow = concat + (size_t)n * (2 * H);
    for (int k = 0; k < 4; ++k) h_s[t + k * 256] = hrow[t + k * 256];
    __syncthreads();

    float cacc[4] = {0.f, 0.f, 0.f, 0.f};
    for (int s = sc * 64; s < sc * 64 + 64; ++s) {
        const float* erow = enc + ((size_t)n * S + s) * H;
        float e[4];
        float local = 0.0f;
        #pragma unroll
        for (int k = 0; k < 4; ++k) {
            e[k] = erow[t + k * 256];
            local += e[k] * h_s[t + k * 256];
        }
        // wave32 reduce, then cross-wave via LDS
        for (int off = 16; off; off >>= 1) local += __shfl_xor(local, off, 32);
        if ((t & 31) == 0) wsum[t >> 5] = local;
        __syncthreads();
        float score = wsum[0] + wsum[1] + wsum[2] + wsum[3] +
                      wsum[4] + wsum[5] + wsum[6] + wsum[7];
        __syncthreads();
        #pragma unroll
        for (int k = 0; k < 4; ++k) cacc[k] += score * e[k];
    }
    float* ctx = concat + (size_t)n * (2 * H) + H;
    #pragma unroll
    for (int k = 0; k < 4; ++k) atomicAdd(&ctx[t + k * 256], cacc[k]);
}

// ---------------------------------------------------------------------------
// logits[n][v] = sum_k concat[n][k] * fc_W[v][k] + fc_b[v]
// One wave per 16-col tile vt (2000 waves). 4 accumulators cover M=64, so
// each fc_W fragment (262 MB total, > L2) is loaded exactly once from HBM
// and reused for 4 WMMAs. concat A-rows are 512 KB -> L2-resident.
// ---------------------------------------------------------------------------
__global__ void k_logits(const float* __restrict__ concat,
                         const float* __restrict__ fcW,
                         const float* __restrict__ fcb,
                         float* __restrict__ out) {
    const int vt    = blockIdx.x * (blockDim.x >> 5) + (threadIdx.x >> 5); // 0..1999
    const int lane  = threadIdx.x & 31;
    const int mrow  = lane & 15;
    const int khalf = (lane >> 4) << 1;

    const float* brow = fcW + (size_t)(vt * 16 + mrow) * (2 * H);
    const float* a0   = concat + (size_t)mrow * (2 * H);   // mt=0 row; others +16*2H

    v8f acc[4] = {v8f{}, v8f{}, v8f{}, v8f{}};

    for (int k = 0; k < 2 * H; k += 4) {
        v2f b;
        b.x = brow[k + khalf]; b.y = brow[k + khalf + 1];
        #pragma unroll
        for (int mt = 0; mt < 4; ++mt) {
            const float* ar = a0 + (size_t)mt * 16 * (2 * H);
            v2f a;
            a.x = ar[k + khalf]; a.y = ar[k + khalf + 1];
            acc[mt] = __builtin_amdgcn_wmma_f32_16x16x4_f32(
                false, a, false, b, (short)0, acc[mt], false, false);
        }
    }

    const int   v    = vt * 16 + (lane & 15);
    const float bias = fcb[v];
    const int   mofs = (lane >> 4) * 8;
    #pragma unroll
    for (int mt = 0; mt < 4; ++mt) {
        #pragma unroll
        for (int r = 0; r < 8; ++r) {
            int row = mt * 16 + r + mofs;
            out[(size_t)row * V + v] = acc[mt][r] + bias;
        }
    }
}

// ---------------------------------------------------------------------------
// In-place log-softmax per row of 32000. One block per row.
// ---------------------------------------------------------------------------
__global__ void k_lsm(float* __restrict__ logits) {
    __shared__ float red[256];
    const int n = blockIdx.x, t = threadIdx.x;
    float* row = logits + (size_t)n * V;

    float m = -1e30f;
    for (int i = t; i < V; i += 256) m = fmaxf(m, row[i]);
    red[t] = m; __syncthreads();
    for (int s = 128; s; s >>= 1) {
        if (t < s) red[t] = fmaxf(red[t], red[t + s]);
        __syncthreads();
    }
    m = red[0]; __syncthreads();

    float sum = 0.0f;
    for (int i = t; i < V; i += 256) sum += __expf(row[i] - m);
    red[t] = sum; __syncthreads();
    for (int s = 128; s; s >>= 1) {
        if (t < s) red[t] += red[t + s];
        __syncthreads();
    }
    float lse = m + logf(red[0]);
    __syncthreads();
    for (int i = t; i < V; i += 256) row[i] = row[i] - lse;
}

// ---------------------------------------------------------------------------
extern "C" void kernel_launch(void* const* d_in, const int* in_sizes, int n_in,
                              void* d_out, int out_size, void* d_ws, size_t ws_size,
                              hipStream_t stream) {
    const int*   prev_outputs = (const int*)  d_in[0];   // (N,1)
    const float* prev_hn      = (const float*)d_in[1];   // (1,N,H)
    const float* prev_cn      = (const float*)d_in[2];   // (1,N,H)
    const float* enc          = (const float*)d_in[3];   // (N,S,H)
    const float* embW         = (const float*)d_in[4];   // (V,E)
    const float* w_ih         = (const float*)d_in[5];   // (4H,E)
    const float* w_hh         = (const float*)d_in[6];   // (4H,H)
    const float* b_ih         = (const float*)d_in[7];   // (4H)
    const float* b_hh         = (const float*)d_in[8];   // (4H)
    const float* fcW          = (const float*)d_in[9];   // (V,2H)
    const float* fcb          = (const float*)d_in[10];  // (V)

    float* out   = (float*)d_out;
    float* logp  = out;                     // N*V
    float* h_out = out + (size_t)NB * V;    // N*H
    float* c_out = h_out + (size_t)NB * H;  // N*H

    float* gates  = (float*)d_ws;                // N * 4H floats (1 MB)
    float* concat = gates + (size_t)NB * 4 * H;  // N * 2H floats (512 KB)

    // concat context half is accumulated with atomics -> must start at zero
    k_zero<<<(NB * 2 * H + 255) / 256, 256, 0, stream>>>(concat, NB * 2 * H);

    // gates GEMM: 256 col tiles, 8 waves/block -> 32 blocks (each wave: M=64)
    k_gates<<<32, 256, 0, stream>>>(prev_outputs, prev_hn, embW,
                                    w_ih, w_hh, b_ih, b_hh, gates);

    // LSTM elementwise: 65536 elements
    k_lstm<<<(NB * H + 255) / 256, 256, 0, stream>>>(gates, prev_cn,
                                                     h_out, c_out, concat);

    // fused attention: grid (n, s-chunk)
    k_attn<<<dim3(NB, S / 64), 256, 0, stream>>>(enc, concat);

    // logits GEMM: 2000 col tiles, 8 waves/block -> 250 blocks (each wave: M=64)
    k_logits<<<250, 256, 0, stream>>>(concat, fcW, fcb, logp);

    // log-softmax in place
    k_lsm<<<NB, 256, 0, stream>>>(logp);
}